// BasicModel_28724741276284
// MI455X (gfx1250) — compile-verified
//
#include <hip/hip_runtime.h>
#include <hip/hip_bf16.h>

typedef __attribute__((ext_vector_type(2))) float v2f;
typedef __attribute__((ext_vector_type(8))) float v8f;

#define EMB_DIM 64
#define SET_SIZE 8

// One wave (32 lanes) per candidate. Rows of the 16x64 matrix M:
//   row 0     : hyperedge_emb[h[e]]
//   rows 1..8 : node_emb[X[e][0..7]]
//   rows 9..15: duplicate of row 0 (padding; their Gram entries are never read)
// G = M * M^T computed with 16 x V_WMMA_F32_16X16X4_F32 (full fp32 precision).
__global__ __launch_bounds__(256) void hypergraph_score_kernel(
    const float* __restrict__ node_emb,
    const float* __restrict__ hyperedge_emb,
    const int*   __restrict__ h,
    const int*   __restrict__ X,
    float*       __restrict__ out,
    int n_cand)
{
    const int lane = threadIdx.x & 31;
    const int wave = threadIdx.x >> 5;
    const int e    = blockIdx.x * (blockDim.x >> 5) + wave;
    if (e >= n_cand) return;              // wave-uniform: EXEC stays all-ones below

    // Row handled by this lane (lane l and lane l+16 share row l, split by K%4).
    const int r = lane & 15;
    const float* rowp;
    if (r >= 1 && r <= 8) {
        rowp = node_emb + (size_t)X[e * SET_SIZE + (r - 1)] * EMB_DIM;
    } else {
        rowp = hyperedge_emb + (size_t)h[e] * EMB_DIM;
    }
    const int ksub = (lane >> 4) << 1;    // lanes 0-15 -> K%4 in {0,1}; lanes 16-31 -> {2,3}

    // A-matrix (== B-matrix) fragments: a[t] covers K = 4t..4t+3 of this lane's row.
    v2f a[16];
#pragma unroll
    for (int t = 0; t < 16; ++t) {
        a[t] = *(const v2f*)(rowp + 4 * t + ksub);
    }

    // Gram accumulation: G += A_t * A_t^T over 16 K-steps of 4.
    v8f c = {};
#pragma unroll
    for (int t = 0; t < 16; ++t) {
        c = __builtin_amdgcn_wmma_f32_16x16x4_f32(
                /*neg_a=*/false, a[t],
                /*neg_b=*/false, a[t],
                /*c_mod=*/(short)0, c,
                /*reuse_a=*/false, /*reuse_b=*/false);
    }

    // C/D layout: VGPR d, lanes 0-15 -> G[d][lane]; lanes 16-31 -> G[d+8][lane-16].
    const float INF = __builtin_inff();
    float m_star = INF;   // min over G[0][1..8]
    float m_pair = INF;   // min over G[1..8][1..8] (diagonal included)
    if (lane >= 1 && lane <= 8) {
        m_star = c[0];                       // G[0][lane]
        float mp = c[1];                     // G[1][lane]
        mp = fminf(mp, c[2]);
        mp = fminf(mp, c[3]);
        mp = fminf(mp, c[4]);
        mp = fminf(mp, c[5]);
        mp = fminf(mp, c[6]);
        mp = fminf(mp, c[7]);                // G[2..7][lane]
        m_pair = mp;
    } else if (lane >= 17 && lane <= 24) {
        m_pair = c[0];                       // G[8][lane-16]
    }

    // wave32 min-reduction
#pragma unroll
    for (int off = 16; off > 0; off >>= 1) {
        m_star = fminf(m_star, __shfl_xor(m_star, off, 32));
        m_pair = fminf(m_pair, __shfl_xor(m_pair, off, 32));
    }

    if (lane == 0) {
        // sigmoid is monotonic: min(sigmoid(x)) == sigmoid(min(x))
        out[2 * e + 0] = 1.0f / (1.0f + __expf(-m_star));
        out[2 * e + 1] = 1.0f / (1.0f + __expf(-m_pair));
    }
}

extern "C" void kernel_launch(void* const* d_in, const int* in_sizes, int n_in,
                              void* d_out, int out_size, void* d_ws, size_t ws_size,
                              hipStream_t stream) {
    const float* node_emb      = (const float*)d_in[0];   // [N_NODES, 64]
    const float* hyperedge_emb = (const float*)d_in[1];   // [N_HYPEREDGES, 64]
    const int*   h             = (const int*)d_in[2];     // [E]
    const int*   X             = (const int*)d_in[3];     // [E, 8]
    float*       out           = (float*)d_out;           // [E, 2]

    const int n_cand = in_sizes[2];

    const int waves_per_block = 256 / 32;                 // 8 candidates per block
    const int grid = (n_cand + waves_per_block - 1) / waves_per_block;

    hypergraph_score_kernel<<<grid, 256, 0, stream>>>(
        node_emb, hyperedge_emb, h, X, out, n_cand);
}